// RoutingCapsule_18262200943506
// MI455X (gfx1250) — compile-verified
//
#include <hip/hip_runtime.h>

// CapsNet dynamic routing on MI455X (gfx1250, wave32).
//
// Roofline: u_hat is 94.4MB -> fits in 192MB L2, so we materialize it once
// (WMMA f32 16x16x4, full-f32 precision matching the reference) and run all
// 3 routing iterations from L2 in a single fused per-batch kernel with the
// routing logits resident in LDS. HBM traffic ~105MB total (~4.5us floor at
// 23.3TB/s); FLOPs (~0.66G) are negligible.

typedef __attribute__((ext_vector_type(2))) float v2f;
typedef __attribute__((ext_vector_type(8))) float v8f;

#define NB    128
#define NIN   1152
#define DIN   8
#define NOUT  10
#define DCAP  16
#define NCOLS 160   // NOUT * DCAP, flattened (j,d)

// ---------------------------------------------------------------------------
// Kernel 1: u_hat[b,i,n] = sum_k W[i,n,k] * x[b,i,k]   (n = j*16+d, k = Din)
// Per i: [128 x 8] @ [8 x 160] GEMM. One block per i (8 waves = 8 batch
// tiles of 16 rows); each wave loops the 10 column tiles, 2 WMMAs each.
//
// V_WMMA_F32_16X16X4_F32 fragment layout (ISA 7.12.2, wave32):
//   A 16x4 : lane l, vgpr v -> M = l%16, K = (l/16)*2 + v   (float2 per half)
//   B 4x16 : lane l, vgpr v -> N = l%16, K = (l/16)*2 + v
//   C 16x16: lane l, vgpr v -> N = l%16, M = v + 8*(l/16)
// ---------------------------------------------------------------------------
__global__ __launch_bounds__(256)
void uhat_wmma_kernel(const float* __restrict__ x,   // [B, NIN, DIN]
                      const float* __restrict__ W,   // [NIN, NOUT, DCAP, DIN]
                      float* __restrict__ uhat)      // [B, NIN, NCOLS]
{
    const int tid   = threadIdx.x;
    const int lane  = tid & 31;
    const int btile = tid >> 5;          // wave id in block = batch tile 0..7
    const int i     = blockIdx.x;        // input capsule index 0..1151

    const int half = lane >> 4;          // 0: lanes 0-15, 1: lanes 16-31
    const int l16  = lane & 15;
    const int k0   = half * 2;           // K base of this lane's fragment

    // A fragments: x[btile*16 + l16][i][k0 .. k0+1] and [k0+4 .. k0+5]
    const float* xrow = x + ((size_t)(btile * 16 + l16) * NIN + i) * DIN;
    const v2f a_lo = *(const v2f*)(xrow + k0);
    const v2f a_hi = *(const v2f*)(xrow + k0 + 4);

    const float* wbase = W + (size_t)i * NCOLS * DIN;
    float* obase = uhat + ((size_t)(btile * 16) * NIN + i) * NCOLS;

    // Warm L2 for the next capsule's weights (emits global_prefetch_b8).
    if (i + 1 < NIN) __builtin_prefetch(wbase + NCOLS * DIN + lane * 4, 0, 1);

#pragma unroll
    for (int nt = 0; nt < NOUT; ++nt) {  // 10 column tiles of 16
        // B fragments: B[k][n] = W[i][nt*16 + l16][k]
        const float* wrow = wbase + (nt * 16 + l16) * DIN;
        const v2f b_lo = *(const v2f*)(wrow + k0);
        const v2f b_hi = *(const v2f*)(wrow + k0 + 4);

        v8f acc = {};
        acc = __builtin_amdgcn_wmma_f32_16x16x4_f32(false, a_lo, false, b_lo,
                                                    (short)0, acc, false, false);
        acc = __builtin_amdgcn_wmma_f32_16x16x4_f32(false, a_hi, false, b_hi,
                                                    (short)0, acc, false, false);

#pragma unroll
        for (int v = 0; v < 8; ++v) {
            const int m = v + 8 * half;                       // batch row in tile
            obase[(size_t)m * (NIN * NCOLS) + nt * 16 + l16] = acc[v];
        }
    }
}

// ---------------------------------------------------------------------------
// Kernel 2: fused dynamic routing, one block per batch element.
// Logits b[i,j] live in LDS across all 3 iterations; u_hat[b] (737KB) is
// re-read from L2. Softmax normalizers (1/sum_j exp) precomputed per row;
// exp is a TRANS op (co-executes with VALU on CDNA5), so recomputing
// c = exp(logit)*inv in the reduction keeps LDS under 64KB.
// ---------------------------------------------------------------------------
__global__ __launch_bounds__(256)
void routing_kernel(const float* __restrict__ uhat,  // [B, NIN, NCOLS]
                    const float* __restrict__ bias,  // [NOUT, DCAP]
                    float* __restrict__ out)         // [B, NOUT, DCAP]
{
    const int b   = blockIdx.x;
    const int tid = threadIdx.x;
    const int g   = tid >> 4;       // 16 i-groups
    const int d   = tid & 15;       // capsule-dim lane

    extern __shared__ float smem[];
    float* blog   = smem;                       // [NIN*NOUT]  = 11520 f
    float* inv_sh = blog + NIN * NOUT;          // [NIN]       =  1152 f
    float* part   = inv_sh + NIN;               // [16*NCOLS]  =  2560 f
    float* v_sh   = part + 16 * NCOLS;          // [NCOLS]     =   160 f
                                                // total 61,568 bytes

    const float* ub = uhat + (size_t)b * NIN * NCOLS;

    for (int t = tid; t < NIN * NOUT; t += 256) blog[t] = 0.0f;
    __syncthreads();

    for (int iter = 0; iter < 3; ++iter) {
        // --- softmax row normalizers (logits start at 0, stay O(1..10)) ---
        for (int i = tid; i < NIN; i += 256) {
            float sum = 0.0f;
#pragma unroll
            for (int j = 0; j < NOUT; ++j) sum += __expf(blog[i * NOUT + j]);
            inv_sh[i] = 1.0f / sum;
        }
        __syncthreads();

        // --- s[j,d] = sum_i c[i,j] * u_hat[i, j*16+d] (i split 16 ways) ---
        float accs[NOUT];
#pragma unroll
        for (int j = 0; j < NOUT; ++j) accs[j] = 0.0f;
        for (int i = g; i < NIN; i += 16) {
            const float invi = inv_sh[i];
#pragma unroll
            for (int j = 0; j < NOUT; ++j) {
                const float c = __expf(blog[i * NOUT + j]) * invi;
                accs[j] = fmaf(c, ub[(size_t)i * NCOLS + j * 16 + d], accs[j]);
            }
        }
#pragma unroll
        for (int j = 0; j < NOUT; ++j) part[g * NCOLS + j * 16 + d] = accs[j];
        __syncthreads();

        // --- reduce partials + bias, then squash (v = |s|^2/(1+|s|^2) * s/|s|)
        if (tid < NCOLS) {
            float s = bias[tid];
#pragma unroll
            for (int g2 = 0; g2 < 16; ++g2) s += part[g2 * NCOLS + tid];
            float sq = s * s;                    // sum over d: 16-wide butterfly
            sq += __shfl_xor(sq, 1, 16);
            sq += __shfl_xor(sq, 2, 16);
            sq += __shfl_xor(sq, 4, 16);
            sq += __shfl_xor(sq, 8, 16);
            const float scale = sq / (1.0f + sq) * rsqrtf(sq + 1e-7f);
            v_sh[tid] = scale * s;
        }
        __syncthreads();

        if (iter == 2) {
            if (tid < NCOLS) out[b * NCOLS + tid] = v_sh[tid];
        } else {
            // --- agreement: blog[i,j] += sum_d u_hat[i,j*16+d] * v[j,d] ---
            for (int i = g; i < NIN; i += 16) {
                float myb = 0.0f;
#pragma unroll
                for (int j = 0; j < NOUT; ++j) {
                    float p = ub[(size_t)i * NCOLS + j * 16 + d] * v_sh[j * 16 + d];
                    p += __shfl_xor(p, 1, 16);
                    p += __shfl_xor(p, 2, 16);
                    p += __shfl_xor(p, 4, 16);
                    p += __shfl_xor(p, 8, 16);
                    if (d == j) myb = p;         // lane j keeps dot_j
                }
                if (d < NOUT) blog[i * NOUT + d] += myb;
            }
            __syncthreads();
        }
    }
}

extern "C" void kernel_launch(void* const* d_in, const int* in_sizes, int n_in,
                              void* d_out, int out_size, void* d_ws, size_t ws_size,
                              hipStream_t stream) {
    (void)in_sizes; (void)n_in; (void)out_size; (void)ws_size;
    const float* x    = (const float*)d_in[0];   // [128,1152,8]
    const float* W    = (const float*)d_in[1];   // [1152,10,16,8]
    const float* bias = (const float*)d_in[2];   // [10,16]
    float* out  = (float*)d_out;                 // [128,10,16]
    float* uhat = (float*)d_ws;                  // [128,1152,160] = 94.4MB, L2-resident

    uhat_wmma_kernel<<<NIN, 256, 0, stream>>>(x, W, uhat);

    const size_t smem = (size_t)(NIN * NOUT + NIN + 16 * NCOLS + NCOLS) * sizeof(float);
    routing_kernel<<<NB, 256, smem, stream>>>(uhat, bias, out);
}